// HighLowFrequencyProcessingBlock_22428319219745
// MI455X (gfx1250) — compile-verified
//
#include <hip/hip_runtime.h>
#include <math.h>

#define NN 4
#define CC 64
#define HH 192
#define WW 192
#define HWP (HH*WW)        // 36864
#define EPSV 1e-5f

typedef float v2f __attribute__((ext_vector_type(2)));
typedef float v8f __attribute__((ext_vector_type(8)));

__device__ __forceinline__ int reflect_idx(int i, int L) {
  if (i < 0) i = -i;
  if (i >= L) i = 2 * L - 2 - i;
  return i;
}

// ---------------------------------------------------------------------------
// 1) Global average pool: one block per (n,c)
// ---------------------------------------------------------------------------
__global__ void gap_kernel(const float* __restrict__ x, float* __restrict__ fp) {
  __shared__ float sm[256];
  int b = blockIdx.x;                      // n*CC + c
  const float* p = x + (size_t)b * HWP;
  float s = 0.f;
  for (int i = threadIdx.x; i < HWP; i += 256) s += p[i];
  sm[threadIdx.x] = s;
  __syncthreads();
  for (int off = 128; off > 0; off >>= 1) {
    if (threadIdx.x < off) sm[threadIdx.x] += sm[threadIdx.x + off];
    __syncthreads();
  }
  if (threadIdx.x == 0) fp[b] = sm[0] * (1.0f / (float)HWP);
}

// ---------------------------------------------------------------------------
// 2) Dynamic filter: fw = fp @ Wf^T, GroupNorm(4 x 144), softmax over 9 taps
//    one block per image n, 576 threads
// ---------------------------------------------------------------------------
__global__ void filter_kernel(const float* __restrict__ fp, const float* __restrict__ wf,
                              const float* __restrict__ gg, const float* __restrict__ gb,
                              float* __restrict__ fsm) {
  __shared__ float sfp[CC];
  __shared__ float sfw[576];
  __shared__ float smv[8];                 // mean[0..3], var[4..7]
  int n = blockIdx.x;
  int j = threadIdx.x;                     // 0..575
  if (j < CC) sfp[j] = fp[n * CC + j];
  __syncthreads();
  float acc = 0.f;
  for (int c = 0; c < CC; ++c) acc += sfp[c] * wf[j * CC + c];
  sfw[j] = acc;
  __syncthreads();
  if (j < 4) {
    float s = 0.f, q = 0.f;
    for (int t = 0; t < 144; ++t) { float v = sfw[j * 144 + t]; s += v; q += v * v; }
    float m = s / 144.f;
    smv[j] = m;
    smv[4 + j] = q / 144.f - m * m;
  }
  __syncthreads();
  {
    int g = j / 144;
    float xn = (sfw[j] - smv[g]) * rsqrtf(smv[4 + g] + EPSV);
    xn = xn * gg[j] + gb[j];
    __syncthreads();
    sfw[j] = xn;
  }
  __syncthreads();
  if (j < CC) {
    float v[9], mx = -1e30f;
    for (int k = 0; k < 9; ++k) { v[k] = sfw[j * 9 + k]; mx = fmaxf(mx, v[k]); }
    float s = 0.f;
    for (int k = 0; k < 9; ++k) { v[k] = expf(v[k] - mx); s += v[k]; }
    float inv = 1.f / s;
    for (int k = 0; k < 9; ++k) fsm[(n * CC + j) * 9 + k] = v[k] * inv;
  }
}

// ---------------------------------------------------------------------------
// 3) low = sum_k fsm*tap(reflect-pad, dilation-2 offsets); high = x - low
// ---------------------------------------------------------------------------
__global__ void lowhigh_kernel(const float* __restrict__ x, const float* __restrict__ fsm,
                               float* __restrict__ low, float* __restrict__ high) {
  int b = blockIdx.x;                      // n*CC*HH
  int h = b % HH;
  int nc = b / HH;
  int w = threadIdx.x;
  const float* xc = x + (size_t)nc * HWP;
  const float* f = fsm + nc * 9;
  float acc = 0.f;
#pragma unroll
  for (int k = 0; k < 9; ++k) {
    int dy = (k / 3) * 2 - 2, dx = (k % 3) * 2 - 2;
    int hh = reflect_idx(h + dy, HH), ww = reflect_idx(w + dx, WW);
    acc += f[k] * xc[hh * WW + ww];
  }
  size_t o = (size_t)nc * HWP + h * WW + w;
  low[o] = acc;
  high[o] = xc[h * WW + w] - acc;
}

// ---------------------------------------------------------------------------
// 4) WMMA 1x1 conv with NB-column register blocking.
//    Each wave: NB 16-pixel columns x all CO/16 M tiles; A fragments loaded
//    once per k-chunk (b64) and reused NB times.
// ---------------------------------------------------------------------------
template <int CI, int CO, int ACT, bool CONCAT, int NB>
__global__ void conv1x1_kernel(const float* __restrict__ X, const float* __restrict__ X2,
                               const float* __restrict__ Wt, const float* __restrict__ Bi,
                               const float* __restrict__ alpha, float* __restrict__ Y) {
  constexpr int MT = CO / 16;
  constexpr int KT = CI / 4;
  int wave = (int)((blockIdx.x * blockDim.x + threadIdx.x) >> 5);
  int waves = NN * HWP / (16 * NB);
  if (wave >= waves) return;
  int lane = threadIdx.x & 31;
  int half = lane >> 4, nl = lane & 15;
  int pg0 = wave * 16 * NB;                // 16*NB divides HWP -> never straddles n
  int n = pg0 / HWP, p0 = pg0 % HWP;

  v8f acc[MT][NB];
  v8f zv = {0.f, 0.f, 0.f, 0.f, 0.f, 0.f, 0.f, 0.f};
#pragma unroll
  for (int m = 0; m < MT; ++m)
#pragma unroll
    for (int q = 0; q < NB; ++q) acc[m][q] = zv;

#pragma unroll
  for (int kt = 0; kt < KT; ++kt) {
    int c0 = kt * 4 + 2 * half;
    v2f a[MT];
#pragma unroll
    for (int m = 0; m < MT; ++m)
      a[m] = *(const v2f*)(Wt + (size_t)(m * 16 + nl) * CI + c0);
#pragma unroll
    for (int q = 0; q < NB; ++q) {
      int p = p0 + q * 16 + nl;
      v2f b;
      if (CONCAT) {
        int c1 = c0 + 1;
        b.x = (c0 < 64) ? X[((size_t)n * 64 + c0) * HWP + p]
                        : X2[((size_t)n * 64 + (c0 - 64)) * HWP + p];
        b.y = (c1 < 64) ? X[((size_t)n * 64 + c1) * HWP + p]
                        : X2[((size_t)n * 64 + (c1 - 64)) * HWP + p];
      } else {
        b.x = X[((size_t)n * CI + c0) * HWP + p];
        b.y = X[((size_t)n * CI + c0 + 1) * HWP + p];
      }
#pragma unroll
      for (int m = 0; m < MT; ++m)
        acc[m][q] = __builtin_amdgcn_wmma_f32_16x16x4_f32(false, a[m], false, b, (short)0,
                                                          acc[m][q], false, false);
    }
  }

  float al = (ACT == 1) ? alpha[0] : 0.f;
#pragma unroll
  for (int m = 0; m < MT; ++m) {
#pragma unroll
    for (int q = 0; q < NB; ++q) {
#pragma unroll
      for (int r = 0; r < 8; ++r) {
        int M = m * 16 + r + 8 * half;
        float v = acc[m][q][r] + Bi[M];
        if (ACT == 1) v = (v >= 0.f) ? v : al * v;
        Y[((size_t)n * CO + M) * HWP + p0 + q * 16 + nl] = v;
      }
    }
  }
}

// ---------------------------------------------------------------------------
// 5) Depthwise 3x3 conv, zero padding, dilation DIL, 16 channels
// ---------------------------------------------------------------------------
template <int DIL>
__global__ void dwconv_kernel(const float* __restrict__ X, const float* __restrict__ Wd,
                              const float* __restrict__ Bi, float* __restrict__ Y) {
  int b = blockIdx.x;                      // n*16*HH
  int h = b % HH;
  int ncc = b / HH;                        // n*16 + c
  int c = ncc % 16;
  int w = threadIdx.x;
  const float* xc = X + (size_t)ncc * HWP;
  float acc = Bi[c];
#pragma unroll
  for (int i = 0; i < 3; ++i) {
    int hh = h + (i - 1) * DIL;
    if (hh < 0 || hh >= HH) continue;
#pragma unroll
    for (int j = 0; j < 3; ++j) {
      int ww = w + (j - 1) * DIL;
      if (ww < 0 || ww >= WW) continue;
      acc += Wd[c * 9 + i * 3 + j] * xc[hh * WW + ww];
    }
  }
  Y[(size_t)ncc * HWP + h * WW + w] = acc;
}

// ---------------------------------------------------------------------------
// 6) GroupNorm stats (sum, sumsq) per contiguous channel group, via atomics
// ---------------------------------------------------------------------------
__global__ void zero_stats_kernel(float* __restrict__ stats) {
  if (threadIdx.x < 64) stats[threadIdx.x] = 0.f;
}

__global__ void stats_kernel(const float* __restrict__ X, float* __restrict__ stats,
                             int cpg, int blocksPerGroup) {
  __shared__ float ss[256], sq[256];
  int gid = blockIdx.x / blocksPerGroup;
  int bip = blockIdx.x % blocksPerGroup;
  size_t elems = (size_t)cpg * HWP;
  const float* p = X + (size_t)gid * elems;
  float s = 0.f, q = 0.f;
  for (size_t i = (size_t)bip * 256 + threadIdx.x; i < elems;
       i += (size_t)blocksPerGroup * 256) {
    float v = p[i];
    s += v;
    q += v * v;
  }
  ss[threadIdx.x] = s;
  sq[threadIdx.x] = q;
  __syncthreads();
  for (int off = 128; off > 0; off >>= 1) {
    if (threadIdx.x < off) {
      ss[threadIdx.x] += ss[threadIdx.x + off];
      sq[threadIdx.x] += sq[threadIdx.x + off];
    }
    __syncthreads();
  }
  if (threadIdx.x == 0) {
    atomicAdd(&stats[gid * 2 + 0], ss[0]);
    atomicAdd(&stats[gid * 2 + 1], sq[0]);
  }
}

// ---------------------------------------------------------------------------
// 7) GroupNorm-normalize + LeakyReLU(0.01)  (high branch, 64ch / 4 groups)
// ---------------------------------------------------------------------------
__global__ void gn_leaky_kernel(const float* __restrict__ X, const float* __restrict__ stats,
                                const float* __restrict__ gg, const float* __restrict__ gb,
                                float* __restrict__ Z) {
  size_t idx = (size_t)blockIdx.x * 256 + threadIdx.x;
  int c = (int)((idx / HWP) % CC);
  int n = (int)(idx / ((size_t)CC * HWP));
  int g = n * 4 + c / 16;
  float cnt = 16.f * (float)HWP;
  float m = stats[g * 2] / cnt;
  float var = stats[g * 2 + 1] / cnt - m * m;
  float z = (X[idx] - m) * rsqrtf(var + EPSV) * gg[c] + gb[c];
  Z[idx] = (z >= 0.f) ? z : 0.01f * z;
}

// ---------------------------------------------------------------------------
// 8) GroupNorm-normalize + exact GELU, in place (fusion, 128ch / 4 groups)
// ---------------------------------------------------------------------------
__global__ void gn_gelu_kernel(float* __restrict__ X, const float* __restrict__ stats,
                               const float* __restrict__ gg, const float* __restrict__ gb) {
  size_t idx = (size_t)blockIdx.x * 256 + threadIdx.x;
  int c = (int)((idx / HWP) % 128);
  int n = (int)(idx / ((size_t)128 * HWP));
  int g = n * 4 + c / 32;
  float cnt = 32.f * (float)HWP;
  float m = stats[g * 2] / cnt;
  float var = stats[g * 2 + 1] / cnt - m * m;
  float z = (X[idx] - m) * rsqrtf(var + EPSV) * gg[c] + gb[c];
  X[idx] = 0.5f * z * (1.0f + erff(z * 0.70710678118654752440f));
}

// ---------------------------------------------------------------------------
// 9a) Repack 3x3 weights OIHW -> [tap(ij)][c-pair][o][2] so each WMMA A
//     fragment is one aligned, lane-coalesced b64 load.
//     Wr[((ij*32 + c/2)*64 + o)*2 + (c&1)] = W3[((o*64+c)*3+i)*3+j]
// ---------------------------------------------------------------------------
__global__ void repack_w3_kernel(const float* __restrict__ W3, float* __restrict__ Wr) {
  int idx = blockIdx.x * 256 + threadIdx.x;     // 0 .. 9*64*64-1
  if (idx >= 9 * 64 * 64) return;
  int s = idx & 1;
  int t = idx >> 1;
  int o = t & 63;
  int u = t >> 6;
  int pr = u & 31;
  int ij = u >> 5;
  int c = pr * 2 + s;
  int i = ij / 3, j = ij % 3;
  Wr[idx] = W3[((o * 64 + c) * 3 + i) * 3 + j];
}

// ---------------------------------------------------------------------------
// 9b) WMMA implicit-GEMM 3x3 conv 64->64, zero pad, + bias + residual.
//     Each wave: 2 x 16-pixel columns (32 px along W), all 4 out-ch tiles.
//     A fragments (b64 from repacked weights) reused across both columns.
// ---------------------------------------------------------------------------
__global__ void conv3_wmma_kernel(const float* __restrict__ Z, const float* __restrict__ Wr,
                                  const float* __restrict__ Bi, const float* __restrict__ Res,
                                  float* __restrict__ Y) {
  int wave = (int)((blockIdx.x * blockDim.x + threadIdx.x) >> 5);
  int waves = NN * HH * (WW / 32);
  if (wave >= waves) return;
  int lane = threadIdx.x & 31;
  int half = lane >> 4, nl = lane & 15;
  int wp = wave % (WW / 32);
  int rem = wave / (WW / 32);
  int h = rem % HH;
  int n = rem / HH;
  int w0 = wp * 32;

  v8f acc[4][2];
  v8f zv = {0.f, 0.f, 0.f, 0.f, 0.f, 0.f, 0.f, 0.f};
#pragma unroll
  for (int m = 0; m < 4; ++m) {
    acc[m][0] = zv;
    acc[m][1] = zv;
  }

#pragma unroll
  for (int i = 0; i < 3; ++i) {
    int hh = h + i - 1;
    if (hh < 0 || hh >= HH) continue;      // wave-uniform: EXEC stays full
    const float* zrow = Z + (size_t)n * CC * HWP + (size_t)hh * WW;
#pragma unroll
    for (int j = 0; j < 3; ++j) {
      int colw0 = w0 + nl + j - 1;
      int colw1 = colw0 + 16;
      bool in0 = (colw0 >= 0) && (colw0 < WW);
      bool in1 = (colw1 >= 0) && (colw1 < WW);
      int cc0 = in0 ? colw0 : 0;
      int cc1 = in1 ? colw1 : 0;
      const float* wtap = Wr + (size_t)(i * 3 + j) * 32 * 64 * 2;
      for (int cb = 0; cb < CC; cb += 4) {
        int pairIdx = (cb >> 1) + half;
        int c0 = cb + 2 * half;
        v2f a[4];
#pragma unroll
        for (int m = 0; m < 4; ++m)
          a[m] = *(const v2f*)(wtap + ((size_t)pairIdx * 64 + m * 16 + nl) * 2);
        v2f b0, b1;
        float b0x = zrow[(size_t)c0 * HWP + cc0];
        float b0y = zrow[(size_t)(c0 + 1) * HWP + cc0];
        float b1x = zrow[(size_t)c0 * HWP + cc1];
        float b1y = zrow[(size_t)(c0 + 1) * HWP + cc1];
        b0.x = in0 ? b0x : 0.f;
        b0.y = in0 ? b0y : 0.f;
        b1.x = in1 ? b1x : 0.f;
        b1.y = in1 ? b1y : 0.f;
#pragma unroll
        for (int m = 0; m < 4; ++m)
          acc[m][0] = __builtin_amdgcn_wmma_f32_16x16x4_f32(false, a[m], false, b0, (short)0,
                                                            acc[m][0], false, false);
#pragma unroll
        for (int m = 0; m < 4; ++m)
          acc[m][1] = __builtin_amdgcn_wmma_f32_16x16x4_f32(false, a[m], false, b1, (short)0,
                                                            acc[m][1], false, false);
      }
    }
  }

#pragma unroll
  for (int q = 0; q < 2; ++q) {
    int p = h * WW + w0 + q * 16 + nl;
#pragma unroll
    for (int m = 0; m < 4; ++m) {
#pragma unroll
      for (int r = 0; r < 8; ++r) {
        int M = m * 16 + r + 8 * half;
        size_t o = ((size_t)n * CC + M) * HWP + p;
        Y[o] = acc[m][q][r] + Bi[M] + Res[o];
      }
    }
  }
}

// ---------------------------------------------------------------------------
extern "C" void kernel_launch(void* const* d_in, const int* in_sizes, int n_in,
                              void* d_out, int out_size, void* d_ws, size_t ws_size,
                              hipStream_t stream) {
  (void)in_sizes; (void)n_in; (void)out_size; (void)ws_size;
  const float* x        = (const float*)d_in[0];
  const float* w_filter = (const float*)d_in[1];
  const float* gnf_g    = (const float*)d_in[2];
  const float* gnf_b    = (const float*)d_in[3];
  const float* lb_w1    = (const float*)d_in[4];
  const float* lb_b1    = (const float*)d_in[5];
  const float* lb_a1    = (const float*)d_in[6];
  const float* lb_dw1   = (const float*)d_in[7];
  const float* lb_db1   = (const float*)d_in[8];
  const float* lb_w2    = (const float*)d_in[9];
  const float* lb_b2    = (const float*)d_in[10];
  const float* lb_a2    = (const float*)d_in[11];
  const float* lb_dw2   = (const float*)d_in[12];
  const float* lb_db2   = (const float*)d_in[13];
  const float* lb_w3    = (const float*)d_in[14];
  const float* lb_b3    = (const float*)d_in[15];
  const float* lb_a3    = (const float*)d_in[16];
  const float* lb_dw3   = (const float*)d_in[17];
  const float* lb_db3   = (const float*)d_in[18];
  const float* lb_w4    = (const float*)d_in[19];
  const float* lb_b4    = (const float*)d_in[20];
  const float* lb_a4    = (const float*)d_in[21];
  const float* lb_w5    = (const float*)d_in[22];
  const float* lb_b5    = (const float*)d_in[23];
  const float* hb_gn_g  = (const float*)d_in[24];
  const float* hb_gn_b  = (const float*)d_in[25];
  const float* hb_w     = (const float*)d_in[26];
  const float* hb_b     = (const float*)d_in[27];
  const float* fu_w1    = (const float*)d_in[28];
  const float* fu_b1    = (const float*)d_in[29];
  const float* fu_gn_g  = (const float*)d_in[30];
  const float* fu_gn_b  = (const float*)d_in[31];
  const float* fu_w2    = (const float*)d_in[32];
  const float* fu_b2    = (const float*)d_in[33];
  float* out = (float*)d_out;

  float* ws    = (float*)d_ws;
  float* fp    = ws;                       // 256
  float* fsm   = ws + 256;                 // 2304
  float* stats = ws + 2560;                // 64
  const size_t T64  = (size_t)NN * CC * HWP;     // 9437184
  const size_t T16  = (size_t)NN * 16 * HWP;     // 2359296
  const size_t T128 = (size_t)NN * 128 * HWP;    // 18874368
  float* lowb = ws + 4096;                 // also reused as z-buffer
  float* highb = lowb + T64;
  float* enhl = highb + T64;
  float* enhh = enhl + T64;
  float* y0 = enhh + T64;
  float* y1 = y0 + T16;
  float* obuf = y1 + T16;                  // 128ch, in-place GN+GELU
  float* wrep = obuf + T128;               // 36864 repacked 3x3 weights

  // conv grid sizes: each block has 8 waves of 32
  const int blocksNB4 = (NN * HWP / (16 * 4)) / 8;   // 288
  const int blocksNB2 = (NN * HWP / (16 * 2)) / 8;   // 576
  const int conv3Blocks = (NN * HH * (WW / 32)) / 8; // 576

  // filter generation
  gap_kernel<<<NN * CC, 256, 0, stream>>>(x, fp);
  filter_kernel<<<NN, 576, 0, stream>>>(fp, w_filter, gnf_g, gnf_b, fsm);
  // low / high split
  lowhigh_kernel<<<NN * CC * HH, WW, 0, stream>>>(x, fsm, lowb, highb);

  // low branch step 1 consumes lowb; afterwards lowb is reused as z
  conv1x1_kernel<64, 16, 1, false, 4><<<blocksNB4, 256, 0, stream>>>(
      lowb, nullptr, lb_w1, lb_b1, lb_a1, y0);

  // high branch: GN stats -> normalize+leaky into lowb (z) -> 3x3 conv + residual
  zero_stats_kernel<<<1, 64, 0, stream>>>(stats);
  stats_kernel<<<16 * 32, 256, 0, stream>>>(highb, stats, 16, 32);
  gn_leaky_kernel<<<(int)(T64 / 256), 256, 0, stream>>>(highb, stats, hb_gn_g, hb_gn_b, lowb);
  repack_w3_kernel<<<(9 * 64 * 64) / 256, 256, 0, stream>>>(hb_w, wrep);
  conv3_wmma_kernel<<<conv3Blocks, 256, 0, stream>>>(lowb, wrep, hb_b, highb, enhh);

  // remaining low branch
  dwconv_kernel<5><<<NN * 16 * HH, WW, 0, stream>>>(y0, lb_dw1, lb_db1, y1);
  conv1x1_kernel<16, 16, 1, false, 4><<<blocksNB4, 256, 0, stream>>>(
      y1, nullptr, lb_w2, lb_b2, lb_a2, y0);
  dwconv_kernel<1><<<NN * 16 * HH, WW, 0, stream>>>(y0, lb_dw2, lb_db2, y1);
  conv1x1_kernel<16, 16, 1, false, 4><<<blocksNB4, 256, 0, stream>>>(
      y1, nullptr, lb_w3, lb_b3, lb_a3, y0);
  dwconv_kernel<3><<<NN * 16 * HH, WW, 0, stream>>>(y0, lb_dw3, lb_db3, y1);
  conv1x1_kernel<16, 16, 1, false, 4><<<blocksNB4, 256, 0, stream>>>(
      y1, nullptr, lb_w4, lb_b4, lb_a4, y0);
  conv1x1_kernel<16, 64, 0, false, 4><<<blocksNB4, 256, 0, stream>>>(
      y0, nullptr, lb_w5, lb_b5, nullptr, enhl);

  // fusion: concat(enh_low, enh_high) -> 1x1 (128->128) -> GN -> GELU -> 1x1 (128->64)
  conv1x1_kernel<128, 128, 0, true, 2><<<blocksNB2, 256, 0, stream>>>(
      enhl, enhh, fu_w1, fu_b1, nullptr, obuf);
  zero_stats_kernel<<<1, 64, 0, stream>>>(stats);
  stats_kernel<<<16 * 32, 256, 0, stream>>>(obuf, stats, 32, 32);
  gn_gelu_kernel<<<(int)(T128 / 256), 256, 0, stream>>>(obuf, stats, fu_gn_g, fu_gn_b);
  conv1x1_kernel<128, 64, 0, false, 4><<<blocksNB4, 256, 0, stream>>>(
      obuf, nullptr, fu_w2, fu_b2, nullptr, out);
}